// MIPSBruteForceTopK_84550726189766
// MI455X (gfx1250) — compile-verified
//
#include <hip/hip_runtime.h>

// ---------------- problem constants ----------------
#define B_Q    512
#define DIM    256
#define NITEMS 500000
#define TOPK   100
#define NCHUNK 64
#define CHUNK  7813              // ceil(500000 / 64)
#define TB     32                // item columns staged per TDM tile
#define NTILES ((CHUNK + TB - 1) / TB)   // 245
#define QT_PER_BLK 2             // query tiles (waves) per block
#define MERGE_N 8192             // next pow2 >= NCHUNK*TOPK = 6400

typedef __attribute__((ext_vector_type(2))) float v2f;
typedef __attribute__((ext_vector_type(8))) float v8f;
typedef __attribute__((ext_vector_type(4))) unsigned int v4u;
typedef __attribute__((ext_vector_type(8))) int v8i;
typedef __attribute__((ext_vector_type(4))) int v4i;

// ---- TDM: load a TB x DIM fp32 tile (cols c0.., all 256 K-rows) into LDS ----
// D# per ISA 8.3/8.4: 2D tensor, dim0 = columns (contiguous, stride 1 elem),
// dim1 = K rows (stride NITEMS elems). tensor_dim0 = NITEMS -> hardware
// zero-fills reads past the right edge of the item matrix.
__device__ inline void tdm_load_tile(const float* gsrc, unsigned lds_bytes) {
  unsigned long long ga = (unsigned long long)(uintptr_t)gsrc;
  v4u g0;
  g0[0] = 1u;                                            // count=1, user mode
  g0[1] = lds_bytes;                                     // lds_addr
  g0[2] = (unsigned)(ga & 0xFFFFFFFFu);                  // global_addr[31:0]
  g0[3] = (unsigned)((ga >> 32) & 0x01FFFFFFu)           // global_addr[56:32]
        | (2u << 30);                                    // type=2 ("image")
  const unsigned long long td0 = NITEMS;                 // tensor_dim0
  const unsigned td1 = DIM;                              // tensor_dim1
  const unsigned long long st0 = NITEMS;                 // tensor_dim0_stride
  v8i g1;
  g1[0] = (int)(2u << 16);                               // data_size=4B, mask=0
  g1[1] = (int)((td0 & 0xFFFFu) << 16);                  // tensor_dim0[15:0]
  g1[2] = (int)(((td0 >> 16) & 0xFFFFu) | ((td1 & 0xFFFFu) << 16));
  g1[3] = (int)(((td1 >> 16) & 0xFFFFu) | ((unsigned)TB << 16));  // tile_dim0
  g1[4] = (int)DIM;                                      // tile_dim1=256, tile_dim2=0
  g1[5] = (int)(st0 & 0xFFFFFFFFu);                      // stride0[31:0]
  g1[6] = (int)((st0 >> 32) & 0xFFFFu);                  // stride0[47:32]
  g1[7] = 0;                                             // dim1_stride unused (2D)
  v4i z4 = {0, 0, 0, 0};
  v8i z8 = {0, 0, 0, 0, 0, 0, 0, 0};
  __builtin_amdgcn_tensor_load_to_lds(g0, g1, z4, z4, z8, 0);
}

// ---------------- kernel 1: fused WMMA GEMM + per-chunk top-K ----------------
// grid = (16, NCHUNK), block = 64 (2 waves; each wave owns one 16-query tile).
// Both waves share a TDM double-buffered 32-col x 256-K item tile in LDS.
__global__ __launch_bounds__(QT_PER_BLK * 32)
void mips_wmma_chunk_topk(const float* __restrict__ Q,
                          const float* __restrict__ IT,
                          float* __restrict__ ws_sc,
                          int*   __restrict__ ws_id) {
  __shared__ float qs[QT_PER_BLK * 16 * DIM];      // 32 KB query tiles
  __shared__ float bt[2][DIM * TB];                // 64 KB double-buffered B tile
  __shared__ float hs[QT_PER_BLK * 16 * TOPK];     // row-heaps: scores
  __shared__ int   hi[QT_PER_BLK * 16 * TOPK];     // row-heaps: item indices
  __shared__ int   hcnt[QT_PER_BLK * 16];

  const int tid   = threadIdx.x;
  const int lane  = tid & 31;
  const int wid   = tid >> 5;                      // wave = query tile
  const int chunk = blockIdx.y;
  const int row0  = blockIdx.x * (QT_PER_BLK * 16);// first query row of block
  const long col_base = (long)chunk * CHUNK;
  long ce = col_base + CHUNK;
  const int col_end = (ce > NITEMS) ? NITEMS : (int)ce;

  // kick off TDM for tile 0 while we stage Q
  if (wid == 0)
    tdm_load_tile(IT + col_base, (unsigned)(uintptr_t)&bt[0][0]);

  for (int i = tid; i < QT_PER_BLK * 16 * DIM; i += QT_PER_BLK * 32)
    qs[i] = Q[(long)(row0 + (i >> 8)) * DIM + (i & 255)];
  if (tid < QT_PER_BLK * 16) hcnt[tid] = 0;

  if (wid == 0) __builtin_amdgcn_s_wait_tensorcnt(0);
  __syncthreads();

  const int rowA  = lane & 15;                     // A-matrix M row of this lane
  const int rsel  = lane >> 4;                     // half-wave select
  const int qbase = wid * 16 * DIM;
  const int hbase = wid * 16;                      // heap row base for this wave

  // inlined min-heap push on this wave's LDS heaps
  auto push = [&](int lrow, float s, int id) {
    float* hrow = &hs[(hbase + lrow) * TOPK];
    int*   irow = &hi[(hbase + lrow) * TOPK];
    int n = hcnt[hbase + lrow];
    if (n < TOPK) {                                // sift-up insert
      int i = n;
      while (i > 0) {
        int p = (i - 1) >> 1;
        float ps = hrow[p];
        if (ps <= s) break;
        hrow[i] = ps; irow[i] = irow[p];
        i = p;
      }
      hrow[i] = s; irow[i] = id;
      hcnt[hbase + lrow] = n + 1;
    } else if (s > hrow[0]) {                      // replace root, sift-down
      int i = 0;
      for (;;) {
        int l = 2 * i + 1, r = l + 1, m = i;
        float ms = s;
        if (l < TOPK) { float ls = hrow[l]; if (ls < ms) { m = l; ms = ls; } }
        if (r < TOPK) { float rs = hrow[r]; if (rs < ms) { m = r; ms = rs; } }
        if (m == i) break;
        hrow[i] = hrow[m]; irow[i] = irow[m];
        i = m;
      }
      hrow[i] = s; irow[i] = id;
    }
  };

  for (int t = 0; t < NTILES; ++t) {
    const int   buf = t & 1;
    const long  c0  = col_base + (long)t * TB;     // first column of this tile
    const float* bl = &bt[buf][0];

    // prefetch next tile via TDM while we compute on this one
    if (wid == 0 && t + 1 < NTILES)
      tdm_load_tile(IT + (c0 + TB), (unsigned)(uintptr_t)&bt[buf ^ 1][0]);

    // two independent accumulator chains (cols c0..+15 and c0+16..+31)
    v8f acc[2] = {{}, {}};
    #pragma unroll 8
    for (int kk = 0; kk < DIM; kk += 4) {
      const int ka = kk + 2 * rsel;
      v2f a, b0, b1;
      a.x  = qs[qbase + rowA * DIM + ka];
      a.y  = qs[qbase + rowA * DIM + ka + 1];
      b0.x = bl[ka * TB + (lane & 15)];
      b0.y = bl[(ka + 1) * TB + (lane & 15)];
      b1.x = bl[ka * TB + 16 + (lane & 15)];
      b1.y = bl[(ka + 1) * TB + 16 + (lane & 15)];
      acc[0] = __builtin_amdgcn_wmma_f32_16x16x4_f32(
          false, a, false, b0, (short)0, acc[0], false, false);
      acc[1] = __builtin_amdgcn_wmma_f32_16x16x4_f32(
          false, a, false, b1, (short)0, acc[1], false, false);
    }

    // threshold-filtered heap insertion (stale root read is over-inclusive -> safe)
    #pragma unroll
    for (int h = 0; h < 2; ++h) {
      const long mycol = c0 + h * 16 + (lane & 15);
      const bool colvalid = (mycol < col_end);
      #pragma unroll
      for (int r = 0; r < 8; ++r) {
        const float s   = acc[h][r];
        const int   row = r + (rsel << 3);
        bool pass = colvalid &&
                    ((hcnt[hbase + row] < TOPK) || (s > hs[(hbase + row) * TOPK]));
        unsigned int m = (unsigned int)__ballot(pass);
        while (m) {
          int src = __ffs((int)m) - 1;
          if (lane == src) push(row, s, (int)mycol);
          m &= m - 1;
          __asm__ __volatile__("" ::: "memory");   // force threshold re-reads
        }
      }
    }

    if (wid == 0 && t + 1 < NTILES) __builtin_amdgcn_s_wait_tensorcnt(0);
    __syncthreads();
  }

  // dump this wave's heaps (unsorted) to workspace: [row][chunk][slot]
  for (int t = lane; t < 16 * TOPK; t += 32) {
    const int lrow = t / TOPK, slot = t % TOPK;
    const long o = ((long)(row0 + wid * 16 + lrow) * NCHUNK + chunk) * TOPK + slot;
    if (slot < hcnt[hbase + lrow]) {
      ws_sc[o] = hs[(hbase + lrow) * TOPK + slot];
      ws_id[o] = hi[(hbase + lrow) * TOPK + slot];
    } else {
      ws_sc[o] = -3.4e38f; ws_id[o] = -1;
    }
  }
}

// ---------------- kernel 2: per-row merge (bitonic sort in LDS) ----------------
__global__ __launch_bounds__(256)
void mips_topk_merge(const float* __restrict__ ws_sc,
                     const int*   __restrict__ ws_id,
                     const long long* __restrict__ item_ids,
                     float* __restrict__ out_logits,
                     long long* __restrict__ out_ids) {
  __shared__ float ss[MERGE_N];
  __shared__ int   si[MERGE_N];
  const int row = blockIdx.x;
  const int tid = threadIdx.x;
  const int ncand = NCHUNK * TOPK;                 // 6400

  for (int i = tid; i < MERGE_N; i += 256) {
    if (i < ncand) {
      const long o = (long)row * ncand + i;
      ss[i] = ws_sc[o];
      si[i] = ws_id[o];
    } else { ss[i] = -3.4e38f; si[i] = -1; }
  }
  __syncthreads();

  for (unsigned k = 2; k <= MERGE_N; k <<= 1) {    // bitonic sort, descending
    for (unsigned j = k >> 1; j > 0; j >>= 1) {
      for (unsigned i = tid; i < MERGE_N; i += 256) {
        const unsigned ixj = i ^ j;
        if (ixj > i) {
          const float a = ss[i], b = ss[ixj];
          const bool dir = ((i & k) == 0);
          if (dir ? (a < b) : (a > b)) {
            ss[i] = b; ss[ixj] = a;
            const int t = si[i]; si[i] = si[ixj]; si[ixj] = t;
          }
        }
      }
      __syncthreads();
    }
  }

  for (int j = tid; j < TOPK; j += 256) {
    out_logits[(long)row * TOPK + j] = ss[j];
    const int id = si[j];
    out_ids[(long)row * TOPK + j] = (id >= 0) ? item_ids[id] : (long long)-1;
  }
}

// ---------------- launch ----------------
extern "C" void kernel_launch(void* const* d_in, const int* in_sizes, int n_in,
                              void* d_out, int out_size, void* d_ws, size_t ws_size,
                              hipStream_t stream) {
  const float*     Q   = (const float*)d_in[0];     // (512, 256) fp32
  const float*     IT  = (const float*)d_in[1];     // (256, 500000) fp32
  const long long* ids = (const long long*)d_in[2]; // (1, 500000) int64
  (void)in_sizes; (void)n_in; (void)out_size; (void)ws_size;

  float* ws_sc = (float*)d_ws;
  int*   ws_id = (int*)((char*)d_ws +
                        (size_t)B_Q * NCHUNK * TOPK * sizeof(float));

  float*     out_logits = (float*)d_out;
  long long* out_ids    = (long long*)(out_logits + (size_t)B_Q * TOPK);

  dim3 g1(B_Q / (QT_PER_BLK * 16), NCHUNK);        // (16, 64), chunk-major L2 reuse
  mips_wmma_chunk_topk<<<g1, QT_PER_BLK * 32, 0, stream>>>(Q, IT, ws_sc, ws_id);
  mips_topk_merge<<<B_Q, 256, 0, stream>>>(ws_sc, ws_id, ids, out_logits, out_ids);
}